// Model_14053132993139
// MI455X (gfx1250) — compile-verified
//
#include <hip/hip_runtime.h>
#include <math.h>

typedef float v2f __attribute__((ext_vector_type(2)));
typedef float v8f __attribute__((ext_vector_type(8)));

#define CLIP_LO 0.01f
#define CLIP_HI 0.99f
#define EPS_Z   1e-9f
#define EPS_LOG 1e-30f

// ---------------------------------------------------------------------------
// Fused classifier: z = clip(softmax(relu(x@W1+b1)@W2+b2)) + eps
// Block: 256 threads (8 waves), computes 32 rows of x against all 512 cols.
// Wave w: rows 16*(w&1).., cols 128*(w>>1).. (8 tiles of 16x16 f32 accum).
// K staged through LDS in chunks of 16 using k-pair-interleaved layout so the
// B operand of V_WMMA_F32_16X16X4_F32 is a single aligned b64 LDS load.
// ---------------------------------------------------------------------------
#define GD 256   // K (= D)
#define GH 512   // hidden
#define GM 32    // rows per block
#define GKC 16   // K chunk

__global__ __launch_bounds__(256)
void gemm_fused(const float* __restrict__ x, const float* __restrict__ W1,
                const float* __restrict__ b1, const float* __restrict__ W2,
                const float* __restrict__ b2, float* __restrict__ z, int N)
{
    __shared__ float As[GM * GKC];              // 32x16 x-tile
    __shared__ float Bsp[(GKC / 2) * GH * 2];   // k-pair interleaved W1 slab (32KB)
    __shared__ float b1s[GH];
    __shared__ float W2s[GH * 2];
    __shared__ float logitS[GM * 2];

    const int tid    = threadIdx.x;
    const int lane   = tid & 31;
    const int wv     = tid >> 5;
    const int mbase  = (wv & 1) * 16;
    const int cbase  = (wv >> 1) * 128;
    const int blockM = blockIdx.x * GM;

    // stage b1 / W2, zero logit accumulators
    for (int i = tid; i < GH / 4; i += 256)
        *(float4*)&b1s[i * 4] = *(const float4*)&b1[i * 4];
    for (int i = tid; i < (GH * 2) / 4; i += 256)
        *(float4*)&W2s[i * 4] = *(const float4*)&W2[i * 4];
    if (tid < GM * 2) logitS[tid] = 0.f;

    v8f acc[8];
    const v8f vzero = {0.f, 0.f, 0.f, 0.f, 0.f, 0.f, 0.f, 0.f};
#pragma unroll
    for (int j = 0; j < 8; ++j) acc[j] = vzero;

    const int aRow  = lane & 15;   // M row within 16-row tile / N col within tile
    const int kHalf = lane >> 4;   // 0: K+{0,1}, 1: K+{2,3}  (ISA 32-bit A layout)

    for (int kc = 0; kc < GD; kc += GKC) {
        __syncthreads();
        // stage As: 32 rows x 16 cols (128 float4)
        if (tid < 128) {
            int r  = tid >> 2;
            int c4 = (tid & 3) * 4;
            int gr = blockM + r; if (gr >= N) gr = N - 1;   // clamp; stores guarded
            float4 v = *(const float4*)&x[(size_t)gr * GD + kc + c4];
            *(float4*)&As[r * GKC + c4] = v;
        }
        // stage Bsp: W1 rows [kc, kc+16) x 512 cols, interleaving k-pairs:
        // Bsp[(k>>1)*GH*2 + c*2 + (k&1)] = W1[kc+k][c]
        for (int g = tid; g < (GKC * GH) / 4; g += 256) {
            int k = g >> 7;
            int c = (g & 127) * 4;
            float4 v = *(const float4*)&W1[(size_t)(kc + k) * GH + c];
            float* dst = &Bsp[((k >> 1) * GH + c) * 2 + (k & 1)];
            dst[0] = v.x; dst[2] = v.y; dst[4] = v.z; dst[6] = v.w;
        }
        __syncthreads();
#pragma unroll
        for (int kk = 0; kk < GKC; kk += 4) {
            // A operand: lane<16 -> K kk+0/1, lane>=16 -> K kk+2/3 (aligned b64)
            v2f a = *(const v2f*)&As[(mbase + aRow) * GKC + kk + kHalf * 2];
            int kp = (kk >> 1) + kHalf;   // k-pair index of (kk + kHalf*2)
#pragma unroll
            for (int j = 0; j < 8; ++j) {
                int col = cbase + j * 16 + aRow;
                v2f b = *(const v2f*)&Bsp[(kp * GH + col) * 2];
                acc[j] = __builtin_amdgcn_wmma_f32_16x16x4_f32(
                    false, a, false, b, (short)0, acc[j], false, false);
            }
        }
    }
    __syncthreads();

    // Epilogue: +b1, relu, project onto W2 columns; C layout: VGPR v holds
    // rows v (lanes 0-15) / v+8 (lanes 16-31), column = tile base + (lane&15).
    float pr0[8], pr1[8];
#pragma unroll
    for (int v = 0; v < 8; ++v) { pr0[v] = 0.f; pr1[v] = 0.f; }
#pragma unroll
    for (int j = 0; j < 8; ++j) {
        int col  = cbase + j * 16 + aRow;
        float bb  = b1s[col];
        float w20 = W2s[col * 2 + 0];
        float w21 = W2s[col * 2 + 1];
#pragma unroll
        for (int v = 0; v < 8; ++v) {
            float h = acc[j][v] + bb;
            h = h > 0.f ? h : 0.f;
            pr0[v] += h * w20;
            pr1[v] += h * w21;
        }
    }
    int rbase = mbase + kHalf * 8;
#pragma unroll
    for (int v = 0; v < 8; ++v) {
        atomicAdd(&logitS[(rbase + v) * 2 + 0], pr0[v]);   // ds_add_f32
        atomicAdd(&logitS[(rbase + v) * 2 + 1], pr1[v]);
    }
    __syncthreads();

    if (tid < GM) {
        int gr = blockM + tid;
        if (gr < N) {
            float l0 = logitS[tid * 2 + 0] + b2[0];
            float l1 = logitS[tid * 2 + 1] + b2[1];
            float m  = fmaxf(l0, l1);
            float e0 = expf(l0 - m), e1 = expf(l1 - m);
            float inv = 1.f / (e0 + e1);
            float p0 = fminf(fmaxf(e0 * inv, CLIP_LO), CLIP_HI) + EPS_Z;
            float p1 = fminf(fmaxf(e1 * inv, CLIP_LO), CLIP_HI) + EPS_Z;
            z[(size_t)gr * 2 + 0] = p0;
            z[(size_t)gr * 2 + 1] = p1;
        }
    }
}

// ---------------------------------------------------------------------------
// BP kernels. lik[e,s,t] = (labels[e]==t) ? q[s] : 1-q[s]; recomputed on the
// fly everywhere (never materialized). Working set is L2-resident.
// ---------------------------------------------------------------------------
__global__ void init_xmsg(const float* __restrict__ z, const int* __restrict__ task_idx,
                          float* __restrict__ xmsg, int E)
{
    int e = blockIdx.x * blockDim.x + threadIdx.x;
    if (e >= E) return;
    int t = task_idx[e];
    xmsg[2 * e + 0] = z[2 * t + 0];
    xmsg[2 * e + 1] = z[2 * t + 1];
}

__global__ void zero_Lj(float* __restrict__ Lj, const int* __restrict__ n_workers,
                        int S, int cap)
{
    long tot = (long)(*n_workers) * S;
    if (tot > cap) tot = cap;
    long stride = (long)gridDim.x * blockDim.x;
    for (long i = blockIdx.x * (long)blockDim.x + threadIdx.x; i < tot; i += stride)
        Lj[i] = 0.f;
}

__global__ void init_Li(const float* __restrict__ z, float* __restrict__ Li, int n2)
{
    int i = blockIdx.x * blockDim.x + threadIdx.x;
    if (i < n2) Li[i] = logf(z[i]);
}

// Pass A: w[e,s] = <x_msg[e], lik[e,s,:]>; Lj[widx,s] += log(w + 1e-30)
template <int SS>
__global__ void bp_passA(const float* __restrict__ xmsg, const int* __restrict__ labels,
                         const int* __restrict__ widx, const float* __restrict__ q,
                         float* __restrict__ Lj, int E, int S)
{
    __shared__ float qs[32];
    if (threadIdx.x < 32) qs[threadIdx.x] = (threadIdx.x < S) ? q[threadIdx.x] : 0.5f;
    __syncthreads();
    int e = blockIdx.x * blockDim.x + threadIdx.x;
    if (e >= E) return;
    int lab = labels[e];
    float x0 = xmsg[2 * e + 0], x1 = xmsg[2 * e + 1];
    float xl = lab ? x1 : x0;
    float xo = lab ? x0 : x1;
    int wbase = widx[e] * S;
    const int sl = SS ? SS : (S < 32 ? S : 32);
#pragma unroll
    for (int s = 0; s < sl; ++s) {
        float qv = qs[s];
        float w  = xl * qv + xo * (1.f - qv);
        atomicAdd(&Lj[wbase + s], logf(w + EPS_LOG));
    }
}

// Pass B: leave-one-out worker msg -> y[e,:]; log_y stored; Li += log_y
template <int SS>
__global__ void bp_passB(const float* __restrict__ xmsg, const int* __restrict__ labels,
                         const int* __restrict__ widx, const int* __restrict__ task_idx,
                         const float* __restrict__ q, const float* __restrict__ Lj,
                         float* __restrict__ Li, float* __restrict__ logy, int E, int S)
{
    __shared__ float qs[32];
    if (threadIdx.x < 32) qs[threadIdx.x] = (threadIdx.x < S) ? q[threadIdx.x] : 0.5f;
    __syncthreads();
    int e = blockIdx.x * blockDim.x + threadIdx.x;
    if (e >= E) return;
    int lab = labels[e];
    float x0 = xmsg[2 * e + 0], x1 = xmsg[2 * e + 1];
    float xl = lab ? x1 : x0;
    float xo = lab ? x0 : x1;
    int wbase = widx[e] * S;
    const int sl = SS ? SS : (S < 32 ? S : 32);
    float ex[SS ? SS : 32];
    float exmax = -3.0e38f;
#pragma unroll
    for (int s = 0; s < sl; ++s) {
        float qv = qs[s];
        float w  = xl * qv + xo * (1.f - qv);
        float lw = logf(w + EPS_LOG);
        float ev = Lj[wbase + s] - lw;       // leave-one-out in log space
        ex[s] = ev;
        exmax = fmaxf(exmax, ev);
    }
    float ylab = 0.f, yoth = 0.f;
#pragma unroll
    for (int s = 0; s < sl; ++s) {
        float wg = expf(ex[s] - exmax);
        ylab += wg * qs[s];
        yoth += wg * (1.f - qs[s]);
    }
    float inv = 1.f / (ylab + yoth);
    float y0 = (lab ? yoth : ylab) * inv;
    float y1 = (lab ? ylab : yoth) * inv;
    float ly0 = logf(y0 + EPS_LOG);
    float ly1 = logf(y1 + EPS_LOG);
    logy[2 * e + 0] = ly0;
    logy[2 * e + 1] = ly1;
    int ti = task_idx[e];
    atomicAdd(&Li[2 * ti + 0], ly0);
    atomicAdd(&Li[2 * ti + 1], ly1);
}

// Pass C: x_msg = softmax(Li[task] - log_y)
__global__ void bp_passC(const int* __restrict__ task_idx, const float* __restrict__ Li,
                         const float* __restrict__ logy, float* __restrict__ xmsg, int E)
{
    int e = blockIdx.x * blockDim.x + threadIdx.x;
    if (e >= E) return;
    int ti = task_idx[e];
    float v0 = Li[2 * ti + 0] - logy[2 * e + 0];
    float v1 = Li[2 * ti + 1] - logy[2 * e + 1];
    float m  = fmaxf(v0, v1);
    float e0 = expf(v0 - m), e1 = expf(v1 - m);
    float inv = 1.f / (e0 + e1);
    xmsg[2 * e + 0] = e0 * inv;
    xmsg[2 * e + 1] = e1 * inv;
}

// loss = -sum_n sum_t softmax(Li[n])[t] * log(z[n,t]) ; two-stage deterministic
#define NB_RED 1024
__global__ void loss_partial(const float* __restrict__ Li, const float* __restrict__ z,
                             int N, float* __restrict__ red)
{
    __shared__ float sm[256];
    float acc = 0.f;
    for (int n = blockIdx.x * blockDim.x + threadIdx.x; n < N;
         n += gridDim.x * blockDim.x) {
        float a = Li[2 * n + 0], b = Li[2 * n + 1];
        float m = fmaxf(a, b);
        float e0 = expf(a - m), e1 = expf(b - m);
        float inv = 1.f / (e0 + e1);
        acc -= (e0 * inv) * logf(z[2 * n + 0]) + (e1 * inv) * logf(z[2 * n + 1]);
    }
    sm[threadIdx.x] = acc;
    __syncthreads();
    for (int off = 128; off > 0; off >>= 1) {
        if (threadIdx.x < off) sm[threadIdx.x] += sm[threadIdx.x + off];
        __syncthreads();
    }
    if (threadIdx.x == 0) red[blockIdx.x] = sm[0];
}

__global__ void loss_final(const float* __restrict__ red, float* __restrict__ out)
{
    __shared__ float sm[256];
    int t = threadIdx.x;
    sm[t] = red[t] + red[t + 256] + red[t + 512] + red[t + 768];
    __syncthreads();
    for (int off = 128; off > 0; off >>= 1) {
        if (t < off) sm[t] += sm[t + off];
        __syncthreads();
    }
    if (t == 0) out[0] = sm[0];
}

// ---------------------------------------------------------------------------
extern "C" void kernel_launch(void* const* d_in, const int* in_sizes, int n_in,
                              void* d_out, int out_size, void* d_ws, size_t ws_size,
                              hipStream_t stream)
{
    const float* x          = (const float*)d_in[0];
    const int*   task_idx   = (const int*)  d_in[1];
    const int*   worker_idx = (const int*)  d_in[2];
    const int*   labels     = (const int*)  d_in[3];
    const int*   n_workers  = (const int*)  d_in[4];   // device scalar
    const float* W1         = (const float*)d_in[5];
    const float* b1         = (const float*)d_in[6];
    const float* W2         = (const float*)d_in[7];
    const float* b2         = (const float*)d_in[8];
    const float* q          = (const float*)d_in[9];
    float* out = (float*)d_out;

    const int H = in_sizes[6];              // 512
    const int D = in_sizes[5] / H;          // 256
    const int N = in_sizes[0] / D;          // 50000
    const int E = in_sizes[1];              // 1,000,000
    const int S = in_sizes[9];              // 10
    const int N_ITERS = 10;
    const int LJ_CAP = 1 << 20;             // floats reserved for Lj (W*S=20000 actual)

    // workspace layout (256B aligned slices)
    char* p = (char*)d_ws;
    auto take = [&](size_t bytes) { char* r = p; p += (bytes + 255) & ~(size_t)255; return r; };
    float* z    = (float*)take((size_t)N * 2 * 4);
    float* Li   = (float*)take((size_t)N * 2 * 4);
    float* xmsg = (float*)take((size_t)E * 2 * 4);
    float* logy = (float*)take((size_t)E * 2 * 4);
    float* Lj   = (float*)take((size_t)LJ_CAP * 4);
    float* red  = (float*)take((size_t)NB_RED * 4);
    (void)ws_size; (void)n_in; (void)out_size;

    const int eblk = (E + 255) / 256;

    // 1. classifier -> z_learning
    gemm_fused<<<(N + GM - 1) / GM, 256, 0, stream>>>(x, W1, b1, W2, b2, z, N);
    // 2. x_msg0 = z[task_idx]
    init_xmsg<<<eblk, 256, 0, stream>>>(z, task_idx, xmsg, E);

    // 3. BP iterations
    for (int it = 0; it < N_ITERS; ++it) {
        zero_Lj<<<128, 256, 0, stream>>>(Lj, n_workers, S, LJ_CAP);
        init_Li<<<(2 * N + 255) / 256, 256, 0, stream>>>(z, Li, 2 * N);
        if (S == 10) {
            bp_passA<10><<<eblk, 256, 0, stream>>>(xmsg, labels, worker_idx, q, Lj, E, S);
            bp_passB<10><<<eblk, 256, 0, stream>>>(xmsg, labels, worker_idx, task_idx,
                                                   q, Lj, Li, logy, E, S);
        } else {
            bp_passA<0><<<eblk, 256, 0, stream>>>(xmsg, labels, worker_idx, q, Lj, E, S);
            bp_passB<0><<<eblk, 256, 0, stream>>>(xmsg, labels, worker_idx, task_idx,
                                                  q, Lj, Li, logy, E, S);
        }
        if (it != N_ITERS - 1)
            bp_passC<<<eblk, 256, 0, stream>>>(task_idx, Li, logy, xmsg, E);
    }

    // 4. loss = -(softmax(Li) * log z).sum()  (deterministic two-stage)
    loss_partial<<<NB_RED, 256, 0, stream>>>(Li, z, N, red);
    loss_final<<<1, 256, 0, stream>>>(red, out);
}